// GIN0_14336600834819
// MI455X (gfx1250) — compile-verified
//
#include <hip/hip_runtime.h>

#define BN_EPS 1e-5f

typedef float v2f __attribute__((ext_vector_type(2)));
typedef float v4f __attribute__((ext_vector_type(4)));
typedef float v8f __attribute__((ext_vector_type(8)));

// ---------------------------------------------------------------- async copy
// CDNA5 async global->LDS copy (VGLOBAL op 98, GV mode), tracked by ASYNCcnt.
// VDST = per-lane LDS byte offset (low 32 bits of generic shared pointer),
// VADDR = 64-bit global address, SADDR = off.
__device__ __forceinline__ void async_copy_b128(const float* g, const float* l) {
  unsigned off = (unsigned)(size_t)l;
  asm volatile("global_load_async_to_lds_b128 %0, %1, off"
               :: "v"(off), "v"(g) : "memory");
}

__device__ __forceinline__ void wait_async0() {
  asm volatile("s_wait_asynccnt 0" ::: "memory");
}

// ---------------------------------------------------------------- utilities
__global__ void copy_kernel(const float* __restrict__ src,
                            float* __restrict__ dst, long long n4) {
  long long i = (long long)blockIdx.x * blockDim.x + threadIdx.x;
  if (i < n4) ((v4f*)dst)[i] = ((const v4f*)src)[i];
}

__global__ void zero_kernel(float* __restrict__ p, int n) {
  int i = blockIdx.x * blockDim.x + threadIdx.x;
  if (i < n) p[i] = 0.f;
}

// ------------------------------------------------- GINE message + scatter-add
// agg must be pre-initialized with h (folds the "x + agg" term of GINEConv).
// One thread per (edge, 4 columns): edge_lin (K=16) computed on the fly.
__global__ void message_kernel(const float* __restrict__ h,      // [N, d]
                               const float* __restrict__ ea,     // [E, 16]
                               const int*  __restrict__ srci,    // [E]
                               const int*  __restrict__ dsti,    // [E]
                               const float* __restrict__ we,     // [16, d]
                               const float* __restrict__ be,     // [d]
                               float* __restrict__ agg,          // [N, d]
                               int E, int dlog2) {
  const int qlog2 = dlog2 - 2;                 // d/4 quads per edge
  const long long total = (long long)E << qlog2;
  long long tid = (long long)blockIdx.x * blockDim.x + threadIdx.x;
  if (tid >= total) return;
  int e  = (int)(tid >> qlog2);
  int c0 = (int)(tid & ((1 << qlog2) - 1)) << 2;
  const float* eap = ea + (long long)e * 16;
  v4f acc = *(const v4f*)&be[c0];
#pragma unroll
  for (int k = 0; k < 16; ++k) {
    float ek = eap[k];
    v4f w = *(const v4f*)&we[(k << dlog2) + c0];
    acc.x = fmaf(ek, w.x, acc.x);
    acc.y = fmaf(ek, w.y, acc.y);
    acc.z = fmaf(ek, w.z, acc.z);
    acc.w = fmaf(ek, w.w, acc.w);
  }
  int s = srci[e];
  v4f hv = *(const v4f*)&h[((long long)s << dlog2) + c0];
  float m0 = hv.x + acc.x; m0 = m0 > 0.f ? m0 : 0.f;
  float m1 = hv.y + acc.y; m1 = m1 > 0.f ? m1 : 0.f;
  float m2 = hv.z + acc.z; m2 = m2 > 0.f ? m2 : 0.f;
  float m3 = hv.w + acc.w; m3 = m3 > 0.f ? m3 : 0.f;
  float* ap = &agg[((long long)dsti[e] << dlog2) + c0];
  atomicAdd(ap + 0, m0);
  atomicAdd(ap + 1, m1);
  atomicAdd(ap + 2, m2);
  atomicAdd(ap + 3, m3);
}

// -------------------------------------------------------- WMMA f32 GEMM
// C[M,Ncol] = epilogue(A[M,K] @ W[K,Ncol] + bias)
// flags bit0: relu; bit1: batchnorm (applied after relu, matching reference).
// One block = 8 waves; block stages the K x 64 weight strip in LDS via the
// CDNA5 async-to-LDS path, then each wave computes a 16x64 output strip with
// 4 WMMA accumulators (A fragment reused across the 4 col tiles).
template <int K>
__global__ void gemm_wmma_lds_kernel(const float* __restrict__ A,
                                     const float* __restrict__ W,
                                     const float* __restrict__ bias,
                                     float* __restrict__ C,
                                     int M, int Ncol, int flags,
                                     const float* __restrict__ bn_g,
                                     const float* __restrict__ bn_b,
                                     const float* __restrict__ bn_m,
                                     const float* __restrict__ bn_v) {
  __shared__ float Wlds[K * 64];

  const int colStrips = Ncol >> 6;
  const int group = blockIdx.x / colStrips;  // 8 row-tiles per block
  const int strip = blockIdx.x % colStrips;
  const int col0  = strip << 6;

  // ---- stage W[0:K, col0:col0+64] into LDS (16B chunks, coalesced) ----
  const int nChunks = K * 16;  // 64 floats per row = 16 x b128
  for (int c = threadIdx.x; c < nChunks; c += blockDim.x) {
    int row  = c >> 4;
    int coff = (c & 15) << 2;
    async_copy_b128(&W[(long long)row * Ncol + col0 + coff],
                    &Wlds[(row << 6) + coff]);
  }
  wait_async0();
  __syncthreads();

  const int lane = threadIdx.x & 31;
  const int wv   = threadIdx.x >> 5;
  const int r  = lane & 15;   // row (A) / col (B,D)
  const int hi = lane >> 4;   // lane-half select
  const int kk = hi << 1;     // k-pair {0,1} vs {2,3}

  const int rowTile = (group << 3) + wv;       // wave-uniform
  if (rowTile < (M >> 4)) {
    const int row0 = rowTile << 4;
    v8f acc0 = {0}, acc1 = {0}, acc2 = {0}, acc3 = {0};
    const float* Ap = A + (long long)(row0 + r) * K + kk;
#pragma unroll 4
    for (int k0 = 0; k0 < K; k0 += 4) {
      v2f a; a.x = Ap[k0]; a.y = Ap[k0 + 1];
      const float* Wp0 = &Wlds[((k0 + kk) << 6) + r];
      const float* Wp1 = Wp0 + 64;
      v2f b0; b0.x = Wp0[0];  b0.y = Wp1[0];
      v2f b1; b1.x = Wp0[16]; b1.y = Wp1[16];
      v2f b2; b2.x = Wp0[32]; b2.y = Wp1[32];
      v2f b3; b3.x = Wp0[48]; b3.y = Wp1[48];
      acc0 = __builtin_amdgcn_wmma_f32_16x16x4_f32(false, a, false, b0, (short)0, acc0, false, false);
      acc1 = __builtin_amdgcn_wmma_f32_16x16x4_f32(false, a, false, b1, (short)0, acc1, false, false);
      acc2 = __builtin_amdgcn_wmma_f32_16x16x4_f32(false, a, false, b2, (short)0, acc2, false, false);
      acc3 = __builtin_amdgcn_wmma_f32_16x16x4_f32(false, a, false, b3, (short)0, acc3, false, false);
    }
    const int rowBase = row0 + (hi << 3);
#pragma unroll
    for (int v = 0; v < 8; ++v) {
      long long rowOff = (long long)(rowBase + v) * Ncol;
#pragma unroll
      for (int j = 0; j < 4; ++j) {
        int col = col0 + (j << 4) + r;
        float val = (j == 0) ? acc0[v] : (j == 1) ? acc1[v] : (j == 2) ? acc2[v] : acc3[v];
        val += bias[col];
        if (flags & 1) val = val > 0.f ? val : 0.f;
        if (flags & 2)
          val = (val - bn_m[col]) * rsqrtf(bn_v[col] + BN_EPS) * bn_g[col] + bn_b[col];
        C[rowOff + col] = val;
      }
    }
  }
}

// ---------------------------------------------------------------- pooling
__global__ void pool_kernel(const float* __restrict__ h,   // [N,64]
                            const int* __restrict__ bid,   // [N]
                            float* __restrict__ pooled,    // [G,64]
                            int N) {
  int tid = blockIdx.x * blockDim.x + threadIdx.x;
  if (tid >= N * 64) return;
  int i = tid >> 6, c = tid & 63;
  atomicAdd(&pooled[(bid[i] << 6) + c], h[tid]);
}

// ---------------------------------------------------------------- head MLP
__global__ void head_kernel(const float* __restrict__ pooled,  // [128,64]
                            const float* __restrict__ f1w,     // [64,16]
                            const float* __restrict__ f1b,     // [16]
                            const float* __restrict__ f2w,     // [16,1]
                            const float* __restrict__ f2b,     // [1]
                            float* __restrict__ out) {
  int g = threadIdx.x;
  if (g >= 128) return;
  const float* p = pooled + g * 64;
  float o = f2b[0];
#pragma unroll
  for (int j = 0; j < 16; ++j) {
    float hs = f1b[j];
    for (int k = 0; k < 64; ++k) hs = fmaf(p[k], f1w[k * 16 + j], hs);
    hs = hs > 0.f ? hs : 0.f;
    o = fmaf(hs, f2w[j], o);
  }
  out[g] = o;
}

static inline unsigned int cdiv_ll(long long a, long long b) {
  return (unsigned int)((a + b - 1) / b);
}

extern "C" void kernel_launch(void* const* d_in, const int* in_sizes, int n_in,
                              void* d_out, int out_size, void* d_ws, size_t ws_size,
                              hipStream_t stream) {
  const float* x    = (const float*)d_in[0];
  const float* ea   = (const float*)d_in[1];
  const int*   ei   = (const int*)d_in[2];
  const int*   bid  = (const int*)d_in[3];
  const float* we1  = (const float*)d_in[4];  const float* be1 = (const float*)d_in[5];
  const float* w1a  = (const float*)d_in[6];  const float* b1a = (const float*)d_in[7];
  const float* w1b  = (const float*)d_in[8];  const float* b1b = (const float*)d_in[9];
  const float* bn1g = (const float*)d_in[10]; const float* bn1b = (const float*)d_in[11];
  const float* bn1m = (const float*)d_in[12]; const float* bn1v = (const float*)d_in[13];
  const float* we2  = (const float*)d_in[14]; const float* be2 = (const float*)d_in[15];
  const float* w2a  = (const float*)d_in[16]; const float* b2a = (const float*)d_in[17];
  const float* w2b  = (const float*)d_in[18]; const float* b2b = (const float*)d_in[19];
  const float* bn2g = (const float*)d_in[20]; const float* bn2b = (const float*)d_in[21];
  const float* bn2m = (const float*)d_in[22]; const float* bn2v = (const float*)d_in[23];
  const float* we3  = (const float*)d_in[24]; const float* be3 = (const float*)d_in[25];
  const float* w3a  = (const float*)d_in[26]; const float* b3a = (const float*)d_in[27];
  const float* w3b  = (const float*)d_in[28]; const float* b3b = (const float*)d_in[29];
  const float* bn3g = (const float*)d_in[30]; const float* bn3b = (const float*)d_in[31];
  const float* bn3m = (const float*)d_in[32]; const float* bn3v = (const float*)d_in[33];
  const float* f1w  = (const float*)d_in[34]; const float* f1b = (const float*)d_in[35];
  const float* f2w  = (const float*)d_in[36]; const float* f2b = (const float*)d_in[37];
  float* out = (float*)d_out;

  const int N = 50000, E = 800000, G = 128;
  const int rowTiles = N / 16;                   // 3125
  const int rowGroups = (rowTiles + 7) / 8;      // 391 (8 waves/block)
  const int* srci = ei;
  const int* dsti = ei + E;

  // workspace: 3 rotating [N,256] buffers + pooled[G,64]
  float* bufA   = (float*)d_ws;
  float* bufB   = bufA + (size_t)N * 256;
  float* bufC   = bufB + (size_t)N * 256;
  float* pooled = bufC + (size_t)N * 256;

  const int TB = 256;  // 8 waves/block (wave32)

  // -------- Layer 1: d_in=64, hid=256, d_out=256 --------
  copy_kernel<<<cdiv_ll((long long)N * 16, TB), TB, 0, stream>>>(x, bufB, (long long)N * 16);
  message_kernel<<<cdiv_ll((long long)E * 16, TB), TB, 0, stream>>>(
      x, ea, srci, dsti, we1, be1, bufB, E, 6);
  gemm_wmma_lds_kernel<64><<<rowGroups * 4, TB, 0, stream>>>(
      bufB, w1a, b1a, bufC, N, 256, /*relu*/1, nullptr, nullptr, nullptr, nullptr);
  gemm_wmma_lds_kernel<256><<<rowGroups * 4, TB, 0, stream>>>(
      bufC, w1b, b1b, bufA, N, 256, /*relu+bn*/3, bn1g, bn1b, bn1m, bn1v);

  // -------- Layer 2: d_in=256, hid=128, d_out=128 --------
  copy_kernel<<<cdiv_ll((long long)N * 64, TB), TB, 0, stream>>>(bufA, bufB, (long long)N * 64);
  message_kernel<<<cdiv_ll((long long)E * 64, TB), TB, 0, stream>>>(
      bufA, ea, srci, dsti, we2, be2, bufB, E, 8);
  gemm_wmma_lds_kernel<256><<<rowGroups * 2, TB, 0, stream>>>(
      bufB, w2a, b2a, bufC, N, 128, 1, nullptr, nullptr, nullptr, nullptr);
  gemm_wmma_lds_kernel<128><<<rowGroups * 2, TB, 0, stream>>>(
      bufC, w2b, b2b, bufA, N, 128, 3, bn2g, bn2b, bn2m, bn2v);

  // -------- Layer 3: d_in=128, hid=64, d_out=64 --------
  copy_kernel<<<cdiv_ll((long long)N * 32, TB), TB, 0, stream>>>(bufA, bufB, (long long)N * 32);
  message_kernel<<<cdiv_ll((long long)E * 32, TB), TB, 0, stream>>>(
      bufA, ea, srci, dsti, we3, be3, bufB, E, 7);
  gemm_wmma_lds_kernel<128><<<rowGroups * 1, TB, 0, stream>>>(
      bufB, w3a, b3a, bufC, N, 64, 1, nullptr, nullptr, nullptr, nullptr);
  gemm_wmma_lds_kernel<64><<<rowGroups * 1, TB, 0, stream>>>(
      bufC, w3b, b3b, bufA, N, 64, 3, bn3g, bn3b, bn3m, bn3v);

  // -------- Pool + head --------
  zero_kernel<<<cdiv_ll(G * 64, TB), TB, 0, stream>>>(pooled, G * 64);
  pool_kernel<<<cdiv_ll((long long)N * 64, TB), TB, 0, stream>>>(bufA, bid, pooled, N);
  head_kernel<<<1, 128, 0, stream>>>(pooled, f1w, f1b, f2w, f2b, out);

  (void)in_sizes; (void)n_in; (void)out_size; (void)ws_size;
}